// HGTLayer_23175643529920
// MI455X (gfx1250) — compile-verified
//
#include <hip/hip_runtime.h>
#include <math.h>
#include <stdint.h>

// Problem constants (match reference)
#define BB   4
#define NN   4096
#define EE   65536
#define HH   8
#define DK_  16
#define NT_  3
#define ET_  6
#define DD   128

typedef float v2f __attribute__((ext_vector_type(2)));
typedef float v8f __attribute__((ext_vector_type(8)));

__device__ __forceinline__ v8f wmma_f32_16x16x4(v2f a, v2f b, v8f c) {
    // D = A(16x4,f32) * B(4x16,f32) + C(16x16,f32)
    return __builtin_amdgcn_wmma_f32_16x16x4_f32(
        /*neg_a=*/false, a, /*neg_b=*/false, b,
        /*c_mod=*/(short)0, c, /*reuse_a=*/false, /*reuse_b=*/false);
}

// ---------------------------------------------------------------------------
// Async copy of a contiguous 8 KB tile (16 rows x 128 f32) global -> LDS using
// GLOBAL_LOAD_ASYNC_TO_LDS_B128 (CDNA5). IOFFSET is added to BOTH the global
// and LDS addresses, so a contiguous copy needs one base per lane + immediates.
// 32 lanes x 16B x 16 slices = 8192 B. Completion tracked on ASYNCcnt.
// ---------------------------------------------------------------------------
__device__ __forceinline__ void async_tile_8k(float* lds_base, const float* gsrc, int lane) {
    unsigned           lds = (unsigned)(uintptr_t)lds_base + (unsigned)lane * 16u;
    unsigned long long ga  = (unsigned long long)(uintptr_t)gsrc + (unsigned long long)lane * 16ull;
#define ASYNC_CP(OFF)                                                            \
    asm volatile("global_load_async_to_lds_b128 %0, %1, off offset:" #OFF        \
                 :: "v"(lds), "v"(ga) : "memory")
    ASYNC_CP(0);    ASYNC_CP(512);  ASYNC_CP(1024); ASYNC_CP(1536);
    ASYNC_CP(2048); ASYNC_CP(2560); ASYNC_CP(3072); ASYNC_CP(3584);
    ASYNC_CP(4096); ASYNC_CP(4608); ASYNC_CP(5120); ASYNC_CP(5632);
    ASYNC_CP(6144); ASYNC_CP(6656); ASYNC_CP(7168); ASYNC_CP(7680);
#undef ASYNC_CP
    asm volatile("s_wait_asynccnt 0x0" ::: "memory");
}

// ---------------------------------------------------------------------------
// Init: zero message accumulator + softmax denominators, set max slots to
// encoded(-inf) for the ordered-uint atomicMax trick.
// ---------------------------------------------------------------------------
__global__ void init_kernel(float* __restrict__ acc,
                            unsigned* __restrict__ maxb,
                            float* __restrict__ den) {
    const int i = blockIdx.x * blockDim.x + threadIdx.x;
    acc[i] = 0.0f;
    if (i < BB * ET_ * HH) {
        maxb[i] = 0x007FFFFFu;   // enc(-inf)
        den[i]  = 0.0f;
    }
}

// ---------------------------------------------------------------------------
// Typed projections: per 16-node tile, K/Q/V = select_by_type( x @ W[t] + b[t] ).
// One wave (32 threads) per tile. f32 WMMA 16x16x4, K accumulated over 32 steps.
// ---------------------------------------------------------------------------
__device__ __forceinline__ void do_proj(const float* __restrict__ xs,
                                        const float* __restrict__ W,
                                        const float* __restrict__ bias,
                                        const int* __restrict__ ty,
                                        int m, int hi, int ncol, int rowbase,
                                        float* __restrict__ Obase) {
#pragma unroll 1
    for (int nt = 0; nt < 8; ++nt) {
        float res[8];
#pragma unroll
        for (int r = 0; r < 8; ++r) res[r] = 0.0f;
#pragma unroll 1
        for (int t = 0; t < NT_; ++t) {
            const float* Wt = W + (size_t)t * DD * DD + nt * 16 + ncol;
            v8f acc = {};
#pragma unroll 4
            for (int k = 0; k < 32; ++k) {
                const int kk = 4 * k + hi;
                v2f a; a.x = xs[m * DD + kk];  a.y = xs[m * DD + kk + 1];
                v2f bfr; bfr.x = Wt[(size_t)kk * DD]; bfr.y = Wt[(size_t)(kk + 1) * DD];
                acc = wmma_f32_16x16x4(a, bfr, acc);
            }
            const float bv = bias[t * DD + nt * 16 + ncol];
#pragma unroll
            for (int r = 0; r < 8; ++r)
                if (ty[r] == t) res[r] = acc[r] + bv;
        }
#pragma unroll
        for (int r = 0; r < 8; ++r)
            Obase[(size_t)(rowbase + r) * DD + nt * 16 + ncol] = res[r];
    }
}

__global__ __launch_bounds__(32)
void proj_kernel(const float* __restrict__ x, const int* __restrict__ node_types,
                 const float* __restrict__ Wk, const float* __restrict__ bk,
                 const float* __restrict__ Wq, const float* __restrict__ bq,
                 const float* __restrict__ Wv, const float* __restrict__ bv,
                 float* __restrict__ Ko, float* __restrict__ Qo, float* __restrict__ Vo) {
    __shared__ float xs[16 * DD];
    const int tile = blockIdx.x;
    const int b  = tile / (NN / 16);
    const int n0 = (tile % (NN / 16)) * 16;
    const int lane = threadIdx.x;

    const float* xb = x + ((size_t)b * NN + n0) * DD;
    async_tile_8k(xs, xb, lane);          // CDNA5 async global -> LDS
    __syncthreads();

    const int m       = lane & 15;
    const int hi      = (lane >> 4) << 1;   // 0 or 2
    const int ncol    = lane & 15;
    const int rowbase = (lane >> 4) << 3;   // 0 or 8

    int ty[8];
#pragma unroll
    for (int r = 0; r < 8; ++r)
        ty[r] = node_types[(size_t)b * NN + n0 + rowbase + r];

    const size_t obase = ((size_t)b * NN + n0) * DD;
    do_proj(xs, Wk, bk, ty, m, hi, ncol, rowbase, Ko + obase);
    do_proj(xs, Wq, bq, ty, m, hi, ncol, rowbase, Qo + obase);
    do_proj(xs, Wv, bv, ty, m, hi, ncol, rowbase, Vo + obase);
}

// ---------------------------------------------------------------------------
// Edge pass 1: scores s = (q^T (k @ W_att[et])) * rel_pri[et,h] / sqrt(DK),
// stored per (b,e,h); per-(b,etype,h) running max via ordered-uint atomicMax.
// One thread per (edge, head). W_att + rel_pri staged in LDS. All K/Q gathers
// are 64B-aligned float4 (b128) loads, L2-resident (K/Q/V = 24 MB << 192 MB).
// ---------------------------------------------------------------------------
__global__ __launch_bounds__(256)
void edge_score_kernel(const int* __restrict__ ei, const int* __restrict__ ety,
                       const unsigned char* __restrict__ emask,
                       const float* __restrict__ Kb, const float* __restrict__ Qb,
                       const float* __restrict__ W_att, const float* __restrict__ rel_pri,
                       float* __restrict__ S, unsigned* __restrict__ maxb) {
    __shared__ float wa[ET_ * DK_ * DK_];
    __shared__ float rp[ET_ * HH];
    const int tid = threadIdx.x;
    for (int i = tid; i < ET_ * DK_ * DK_; i += blockDim.x) wa[i] = W_att[i];
    for (int i = tid; i < ET_ * HH;        i += blockDim.x) rp[i] = rel_pri[i];
    __syncthreads();

    const int g = blockIdx.x * blockDim.x + tid;      // < B*E*H = 2^21
    const int h = g & (HH - 1);
    const int e = (g >> 3) & (EE - 1);
    const int b = g >> 19;

    const int src = ei[((size_t)b * 2 + 0) * EE + e];
    const int tgt = ei[((size_t)b * 2 + 1) * EE + e];
    const int et  = ety[(size_t)b * EE + e];
    const bool valid = emask[(size_t)b * EE + e] != 0;

    const float4* k4 = (const float4*)(Kb + (((size_t)b * NN + src) * HH + h) * DK_);
    const float4* q4 = (const float4*)(Qb + (((size_t)b * NN + tgt) * HH + h) * DK_);
    float kv[DK_], qv[DK_];
#pragma unroll
    for (int i = 0; i < 4; ++i) {
        const float4 kq = k4[i]; const float4 qq = q4[i];
        kv[4*i+0] = kq.x; kv[4*i+1] = kq.y; kv[4*i+2] = kq.z; kv[4*i+3] = kq.w;
        qv[4*i+0] = qq.x; qv[4*i+1] = qq.y; qv[4*i+2] = qq.z; qv[4*i+3] = qq.w;
    }

    const float* Wa = wa + et * DK_ * DK_;
    float s = 0.0f;
#pragma unroll 4
    for (int d = 0; d < DK_; ++d) {
        float r = 0.0f;
#pragma unroll
        for (int f = 0; f < DK_; ++f) r += Wa[d * DK_ + f] * qv[f];
        s += kv[d] * r;
    }
    s *= rp[et * HH + h] * 0.25f;   // 1/sqrt(16)

    S[g] = valid ? s : -INFINITY;
    if (valid) {
        unsigned u = __float_as_uint(s);
        u = (u & 0x80000000u) ? ~u : (u | 0x80000000u);
        atomicMax(&maxb[((size_t)b * ET_ + et) * HH + h], u);
    }
}

// ---------------------------------------------------------------------------
// Edge pass 2: e = exp(s - max); per-(b,etype,h) denominator via atomicAdd.
// ---------------------------------------------------------------------------
__global__ __launch_bounds__(256)
void edge_exp_kernel(const int* __restrict__ ety, float* __restrict__ S,
                     const unsigned* __restrict__ maxb, float* __restrict__ den) {
    const int g = blockIdx.x * blockDim.x + threadIdx.x;
    const int h = g & (HH - 1);
    const int e = (g >> 3) & (EE - 1);
    const int b = g >> 19;
    const int et = ety[(size_t)b * EE + e];

    const float s = S[g];
    unsigned u = maxb[((size_t)b * ET_ + et) * HH + h];
    const unsigned dec = (u & 0x80000000u) ? (u & 0x7FFFFFFFu) : ~u;
    float mx = __uint_as_float(dec);
    if (!(mx > -1e38f)) mx = 0.0f;            // mirror reference: -inf max -> 0

    const float ev = (s > -1e38f) ? __expf(s - mx) : 0.0f;
    S[g] = ev;
    if (ev > 0.0f) atomicAdd(&den[((size_t)b * ET_ + et) * HH + h], ev);
}

// ---------------------------------------------------------------------------
// Edge pass 3: att = e/denom; v_rel = src_v @ W_msg[et]; scatter att*v_rel
// into accumulator at target node (16 float atomics per (edge,head)).
// ---------------------------------------------------------------------------
__global__ __launch_bounds__(256)
void edge_scatter_kernel(const int* __restrict__ ei, const int* __restrict__ ety,
                         const float* __restrict__ Vb, const float* __restrict__ W_msg,
                         const float* __restrict__ S, const float* __restrict__ den,
                         float* __restrict__ acc) {
    __shared__ float wm[ET_ * DK_ * DK_];
    const int tid = threadIdx.x;
    for (int i = tid; i < ET_ * DK_ * DK_; i += blockDim.x) wm[i] = W_msg[i];
    __syncthreads();

    const int g = blockIdx.x * blockDim.x + tid;
    const int h = g & (HH - 1);
    const int e = (g >> 3) & (EE - 1);
    const int b = g >> 19;

    const float ev = S[g];
    if (!(ev > 0.0f)) return;

    const int src = ei[((size_t)b * 2 + 0) * EE + e];
    const int tgt = ei[((size_t)b * 2 + 1) * EE + e];
    const int et  = ety[(size_t)b * EE + e];
    const float att = ev / fmaxf(den[((size_t)b * ET_ + et) * HH + h], 1e-20f);

    const float4* v4 = (const float4*)(Vb + (((size_t)b * NN + src) * HH + h) * DK_);
    float vv[DK_];
#pragma unroll
    for (int i = 0; i < 4; ++i) {
        const float4 vq = v4[i];
        vv[4*i+0] = vq.x; vv[4*i+1] = vq.y; vv[4*i+2] = vq.z; vv[4*i+3] = vq.w;
    }

    const float* Wm = wm + et * DK_ * DK_;
    float* o = acc + (((size_t)b * NN + tgt) * HH + h) * DK_;
#pragma unroll 4
    for (int f = 0; f < DK_; ++f) {
        float r = 0.0f;
#pragma unroll
        for (int d = 0; d < DK_; ++d) r += vv[d] * Wm[d * DK_ + f];
        atomicAdd(&o[f], att * r);
    }
}

// ---------------------------------------------------------------------------
// Output projection (WMMA f32) + bias + residual + LayerNorm + node mask.
// One wave per 16-node tile; accumulator tile staged via async LDS copy.
// ---------------------------------------------------------------------------
__global__ __launch_bounds__(32)
void outln_kernel(const float* __restrict__ acc, const float* __restrict__ x,
                  const float* __restrict__ Wout, const float* __restrict__ bout,
                  const float* __restrict__ ln_g, const float* __restrict__ ln_b,
                  const float* __restrict__ node_mask, float* __restrict__ out) {
    __shared__ float as[16 * DD];
    __shared__ float ys[16 * DD];
    const int tile = blockIdx.x;
    const int b  = tile / (NN / 16);
    const int n0 = (tile % (NN / 16)) * 16;
    const int lane = threadIdx.x;

    const float* ab = acc + ((size_t)b * NN + n0) * DD;
    async_tile_8k(as, ab, lane);          // CDNA5 async global -> LDS
    __syncthreads();

    const int m       = lane & 15;
    const int hi      = (lane >> 4) << 1;
    const int ncol    = lane & 15;
    const int rowbase = (lane >> 4) << 3;

#pragma unroll 1
    for (int nt = 0; nt < 8; ++nt) {
        v8f d = {};
        const float* Wt = Wout + nt * 16 + ncol;
#pragma unroll 4
        for (int k = 0; k < 32; ++k) {
            const int kk = 4 * k + hi;
            v2f a; a.x = as[m * DD + kk];  a.y = as[m * DD + kk + 1];
            v2f bfr; bfr.x = Wt[(size_t)kk * DD]; bfr.y = Wt[(size_t)(kk + 1) * DD];
            d = wmma_f32_16x16x4(a, bfr, d);
        }
        const int col = nt * 16 + ncol;
        const float bias = bout[col];
#pragma unroll
        for (int r = 0; r < 8; ++r) {
            const int row = rowbase + r;
            ys[row * DD + col] = d[r] + bias + x[((size_t)b * NN + n0 + row) * DD + col];
        }
    }
    __syncthreads();

    if (lane < 16) {
        const int row = lane;
        float mu = 0.0f;
        for (int c = 0; c < DD; ++c) mu += ys[row * DD + c];
        mu *= (1.0f / DD);
        float var = 0.0f;
        for (int c = 0; c < DD; ++c) { const float dd = ys[row * DD + c] - mu; var += dd * dd; }
        var *= (1.0f / DD);
        const float inv = rsqrtf(var + 1e-5f);
        const float msk = node_mask[(size_t)b * NN + n0 + row];
        float* orow = out + ((size_t)b * NN + n0 + row) * DD;
        for (int c = 0; c < DD; ++c)
            orow[c] = ((ys[row * DD + c] - mu) * inv * ln_g[c] + ln_b[c]) * msk;
    }
}

// ---------------------------------------------------------------------------
extern "C" void kernel_launch(void* const* d_in, const int* in_sizes, int n_in,
                              void* d_out, int out_size, void* d_ws, size_t ws_size,
                              hipStream_t stream) {
    const float* x      = (const float*)d_in[0];
    const int*   ei     = (const int*)d_in[1];
    const int*   ntypes = (const int*)d_in[2];
    const int*   etypes = (const int*)d_in[3];
    const float* nmask  = (const float*)d_in[4];
    const unsigned char* emask = (const unsigned char*)d_in[5];
    const float* Wk     = (const float*)d_in[6];
    const float* bk     = (const float*)d_in[7];
    const float* Wq     = (const float*)d_in[8];
    const float* bq     = (const float*)d_in[9];
    const float* Wv     = (const float*)d_in[10];
    const float* bv     = (const float*)d_in[11];
    const float* relpri = (const float*)d_in[12];
    const float* W_att  = (const float*)d_in[13];
    const float* W_msg  = (const float*)d_in[14];
    const float* W_out  = (const float*)d_in[15];
    const float* b_out  = (const float*)d_in[16];
    const float* ln_g   = (const float*)d_in[17];
    const float* ln_b   = (const float*)d_in[18];
    float* out = (float*)d_out;

    float* ws = (float*)d_ws;
    const size_t SZ = (size_t)BB * NN * DD;   // 2,097,152 floats (== B*E*H)
    float*    K    = ws;
    float*    Q    = ws + SZ;
    float*    V    = ws + 2 * SZ;
    float*    S    = ws + 3 * SZ;
    float*    ACC  = ws + 4 * SZ;
    unsigned* MAXB = (unsigned*)(ws + 5 * SZ);
    float*    DEN  = ws + 5 * SZ + 256;       // 192 used, padded

    const int nEdgeThreads = BB * EE * HH;    // 2,097,152

    init_kernel<<<(int)(SZ / 256), 256, 0, stream>>>(ACC, MAXB, DEN);
    proj_kernel<<<BB * (NN / 16), 32, 0, stream>>>(x, ntypes, Wk, bk, Wq, bq, Wv, bv, K, Q, V);
    edge_score_kernel<<<nEdgeThreads / 256, 256, 0, stream>>>(ei, etypes, emask, K, Q, W_att, relpri, S, MAXB);
    edge_exp_kernel<<<nEdgeThreads / 256, 256, 0, stream>>>(etypes, S, MAXB, DEN);
    edge_scatter_kernel<<<nEdgeThreads / 256, 256, 0, stream>>>(ei, etypes, V, W_msg, S, DEN, ACC);
    outln_kernel<<<BB * (NN / 16), 32, 0, stream>>>(ACC, x, W_out, b_out, ln_g, ln_b, nmask, out);
}